// Policy_11484742550172
// MI455X (gfx1250) — compile-verified
//
#include <hip/hip_runtime.h>
#include <hip/hip_bf16.h>
#include <math.h>

// CDNA5 (gfx1250) wave32 WMMA fragment types
typedef __attribute__((ext_vector_type(2))) float v2f;
typedef __attribute__((ext_vector_type(8))) float v8f;

// Network dims
#define C1 8
#define C2 16
#define C3 32
#define C4 32
#define FC1_OUT 128
#define FC1_IN  960   // 32 channels x 30 length
#define SEGLEN  30    // fc1 weight run collapsing to one c4 channel

__global__ __launch_bounds__(1024)
void policy_fused_kernel(const float* __restrict__ conv1_b,
                         const float* __restrict__ conv2_w,
                         const float* __restrict__ conv2_b,
                         const float* __restrict__ conv3_w,
                         const float* __restrict__ conv3_b,
                         const float* __restrict__ conv4_w,
                         const float* __restrict__ conv4_b,
                         const float* __restrict__ fc1_w,
                         const float* __restrict__ fc1_b,
                         const float* __restrict__ fc2_w,
                         const float* __restrict__ fc2_b,
                         float* __restrict__ out)
{
    __shared__ float s_c1[C1];
    __shared__ float s_c2[C2];
    __shared__ float s_c3[C3];
    __shared__ float s_c4[C4];
    __shared__ float s_wsum[FC1_OUT * 32];   // 128 x 32, 16 KB
    __shared__ float s_y[FC1_OUT];
    __shared__ float s_z[2];

    const int tid = threadIdx.x;

    // ---- Phase A: conv chain on the (provably zero) input -> per-channel constants
    if (tid < C1) s_c1[tid] = fmaxf(conv1_b[tid], 0.0f);
    __syncthreads();

    if (tid < C2) {
        float acc = conv2_b[tid];
        #pragma unroll
        for (int i = 0; i < C1; ++i)
            acc += (conv2_w[tid * (C1 * 2) + i * 2] + conv2_w[tid * (C1 * 2) + i * 2 + 1]) * s_c1[i];
        s_c2[tid] = fmaxf(acc, 0.0f);
    }
    __syncthreads();

    if (tid < C3) {
        float acc = conv3_b[tid];
        #pragma unroll
        for (int i = 0; i < C2; ++i)
            acc += (conv3_w[tid * (C2 * 2) + i * 2] + conv3_w[tid * (C2 * 2) + i * 2 + 1]) * s_c2[i];
        s_c3[tid] = fmaxf(acc, 0.0f);
    }
    __syncthreads();

    if (tid < C4) {
        float acc = conv4_b[tid];
        #pragma unroll
        for (int i = 0; i < C3; ++i)
            acc += conv4_w[tid * C3 + i] * s_c3[i];
        s_c4[tid] = fmaxf(acc, 0.0f);
    }
    // visibility of s_c4 is covered by the barrier at the end of Phase B

    // ---- Phase B: collapse fc1_w [128 x 960] -> Wsum [128 x 32]
    // Each entry sums 30 contiguous floats (8-byte aligned run -> float2 loads).
    // 4096 entries over 1024 threads = 4 per thread; 32 waves of MLP for max MLP/BW.
    for (int e = tid; e < FC1_OUT * 32; e += 1024) {
        const int j = e >> 5;       // output row 0..127
        const int o = e & 31;       // c4 channel 0..31
        const float2* q = reinterpret_cast<const float2*>(fc1_w + j * FC1_IN + o * SEGLEN);
        float s0 = 0.0f, s1 = 0.0f;
        #pragma unroll
        for (int i = 0; i < SEGLEN / 2; ++i) {
            float2 t = q[i];
            s0 += t.x;
            s1 += t.y;
        }
        s_wsum[e] = s0 + s1;        // s_wsum[j*32 + o]
    }
    __syncthreads();

    // ---- Phase WMMA: y = relu(Wsum[128x32] @ c4[32] + fc1_b)  via V_WMMA_F32_16X16X4_F32
    // Wave w owns rows 16w..16w+15. K stepped 4 per WMMA, 8 chained accumulations.
    // A (16x4, f32): lanes 0-15 = M rows with K={4s,4s+1}; lanes 16-31 same rows, K={4s+2,4s+3}.
    // B (4x16): broadcast c4[k] across all N columns -> every D column holds the matvec.
    const int wave = tid >> 5;
    const int lane = tid & 31;
    if (wave < 8) {
        const int m    = lane & 15;
        const int half = lane >> 4;          // 0: K pair {4s,4s+1}, 1: {4s+2,4s+3}
        const int row  = wave * 16 + m;

        v8f acc = {0.0f, 0.0f, 0.0f, 0.0f, 0.0f, 0.0f, 0.0f, 0.0f};
        #pragma unroll
        for (int s = 0; s < 8; ++s) {
            const int k0 = 4 * s + 2 * half;
            v2f a, b;
            a.x = s_wsum[row * 32 + k0];
            a.y = s_wsum[row * 32 + k0 + 1];
            b.x = s_c4[k0];
            b.y = s_c4[k0 + 1];
            // 8 args: (neg_a, A, neg_b, B, c_mod, C, reuse_a, reuse_b)
            acc = __builtin_amdgcn_wmma_f32_16x16x4_f32(
                false, a, false, b, (short)0, acc, false, false);
        }
        // D layout: VGPR r <-> M=r (lanes 0-15) / M=8+r (lanes 16-31); column N = lane%16.
        // Lanes 0 and 16 hold column 0 for rows base..base+7.
        if ((lane & 15) == 0) {
            const int base = wave * 16 + 8 * half;
            #pragma unroll
            for (int r = 0; r < 8; ++r)
                s_y[base + r] = fmaxf(acc[r] + fc1_b[base + r], 0.0f);
        }
    }
    __syncthreads();

    // ---- Phase C: fc2 (2x128 matvec) + softmax
    if (tid < 2) {
        float z = fc2_b[tid];
        #pragma unroll 8
        for (int j = 0; j < FC1_OUT; ++j)
            z += fc2_w[tid * FC1_OUT + j] * s_y[j];
        s_z[tid] = z;
    }
    __syncthreads();

    if (tid == 0) {
        const float z0 = s_z[0], z1 = s_z[1];
        const float mx = fmaxf(z0, z1);
        const float e0 = expf(z0 - mx);
        const float e1 = expf(z1 - mx);
        const float inv = 1.0f / (e0 + e1);
        out[0] = e0 * inv;
        out[1] = e1 * inv;
    }
}

extern "C" void kernel_launch(void* const* d_in, const int* in_sizes, int n_in,
                              void* d_out, int out_size, void* d_ws, size_t ws_size,
                              hipStream_t stream) {
    // setup_inputs() order:
    // 0:x 1:conv1_w 2:conv1_b 3:conv2_w 4:conv2_b 5:conv3_w 6:conv3_b
    // 7:conv4_w 8:conv4_b 9:fc1_w 10:fc1_b 11:fc2_w 12:fc2_b
    // (x and conv1_w are mathematically unused: the reference's pad trick makes
    //  the conv input identically zero.)
    const float* conv1_b = (const float*)d_in[2];
    const float* conv2_w = (const float*)d_in[3];
    const float* conv2_b = (const float*)d_in[4];
    const float* conv3_w = (const float*)d_in[5];
    const float* conv3_b = (const float*)d_in[6];
    const float* conv4_w = (const float*)d_in[7];
    const float* conv4_b = (const float*)d_in[8];
    const float* fc1_w   = (const float*)d_in[9];
    const float* fc1_b   = (const float*)d_in[10];
    const float* fc2_w   = (const float*)d_in[11];
    const float* fc2_b   = (const float*)d_in[12];
    float* out = (float*)d_out;

    policy_fused_kernel<<<1, 1024, 0, stream>>>(
        conv1_b, conv2_w, conv2_b, conv3_w, conv3_b, conv4_w, conv4_b,
        fc1_w, fc1_b, fc2_w, fc2_b, out);
}